// RejectionSampler_5085241278555
// MI455X (gfx1250) — compile-verified
//
#include <hip/hip_runtime.h>

// Problem constants (fixed by the reference).
#define B_ 128
#define L_ 8
#define V_ 32000
constexpr int CH    = V_ / 4;          // 8000 float4 chunks per row
constexpr int WAVES = 8;               // 256 threads / wave32
constexpr int CPW   = CH / WAVES;      // 1000 chunks per wave
constexpr int ITERS = CPW / 32;        // 31 full async tiles per wave
constexpr int TAIL  = CPW - ITERS * 32;// 8 remainder chunks per wave
constexpr int DEPTH = 4;               // async pipeline depth (tiles in flight)

// ---------------- CDNA5 async global->LDS helpers (inline asm) ----------------
__device__ __forceinline__ void async_ld_b128(unsigned lds_byte_off,
                                              unsigned voff_bytes,
                                              unsigned long long sbase) {
  // GVS form: mem_addr = SGPR_u64 + VGPR_i32 ; LDS dest addr in VDST vgpr.
  asm volatile("global_load_async_to_lds_b128 %0, %1, %2"
               :: "v"(lds_byte_off), "v"(voff_bytes), "s"(sbase) : "memory");
}
__device__ __forceinline__ void wait_dscnt0() { asm volatile("s_wait_dscnt 0" ::: "memory"); }
// Wait until tile i is complete. Async loads complete in order; each tile = 2 ops.
// remaining = number of *younger* tiles legitimately still in flight.
__device__ __forceinline__ void wait_async_tiles(int remaining) {
  if (remaining >= 3)      asm volatile("s_wait_asynccnt 6" ::: "memory");
  else if (remaining == 2) asm volatile("s_wait_asynccnt 4" ::: "memory");
  else if (remaining == 1) asm volatile("s_wait_asynccnt 2" ::: "memory");
  else                     asm volatile("s_wait_asynccnt 0" ::: "memory");
}

// argmax merge, first-occurrence (smallest index) tie-break, matching jnp.argmax
__device__ __forceinline__ void upd(float v, int idx, float& best, int& bidx) {
  if (v > best || (v == best && idx < bidx)) { best = v; bidx = idx; }
}

__global__ __launch_bounds__(256) void row_reduce_kernel(
    const float* __restrict__ draft, const float* __restrict__ target,
    const float* __restrict__ q, const int* __restrict__ dids,
    int* __restrict__ targ_arg, int* __restrict__ recov,
    float* __restrict__ pd, float* __restrict__ pt)
{
  const int row = blockIdx.x;            // b*L + l
  const int b   = row >> 3;
  const int tid = threadIdx.x;
  const int w   = tid >> 5;              // wave id (wave32)
  const int ln  = tid & 31;              // lane id

  const float*  trow = target + (size_t)row * V_;
  const float*  drow = draft  + (size_t)row * V_;
  const float4* q4   = (const float4*)(q + (size_t)b * V_);
  const float4* t4   = (const float4*)trow;
  const float4* d4   = (const float4*)drow;

  // wave-private 4-deep circular buffers: [wave][buf][tensor(t/d)][lane] -> 32 KB
  __shared__ float4 smem[WAVES][DEPTH][2][32];
  __shared__ float  s_val[2][WAVES];
  __shared__ int    s_idx[2][WAVES];

  const unsigned long long tbase = (unsigned long long)trow;
  const unsigned long long dbase = (unsigned long long)drow;

  float bt = -__builtin_inff(), br = -__builtin_inff();
  int   it = 0x7fffffff,        ir = 0x7fffffff;

  auto lds_off = [&](int buf, int which) -> unsigned {
    // low 32 bits of the flat address of an LDS object == LDS byte address
    return (unsigned)(unsigned long long)(const void*)&smem[w][buf][which][ln];
  };
  auto issue = [&](int i) {
    wait_dscnt0();  // ds_loads from the buffer being recycled must land before DMA overwrites it
    const int buf = i & (DEPTH - 1);
    unsigned voff = (unsigned)((w * CPW + i * 32 + ln) * 16);
    async_ld_b128(lds_off(buf, 0), voff, tbase);
    async_ld_b128(lds_off(buf, 1), voff, dbase);
  };
  auto proc = [&](const float4& tv, const float4& dv, const float4& qv, int vb) {
    upd(tv.x, vb + 0, bt, it); upd(tv.y, vb + 1, bt, it);
    upd(tv.z, vb + 2, bt, it); upd(tv.w, vb + 3, bt, it);
    float rx = fmaxf(tv.x - dv.x, 0.f) * __builtin_amdgcn_rcpf(qv.x);
    float ry = fmaxf(tv.y - dv.y, 0.f) * __builtin_amdgcn_rcpf(qv.y);
    float rz = fmaxf(tv.z - dv.z, 0.f) * __builtin_amdgcn_rcpf(qv.z);
    float rw = fmaxf(tv.w - dv.w, 0.f) * __builtin_amdgcn_rcpf(qv.w);
    upd(rx, vb + 0, br, ir); upd(ry, vb + 1, br, ir);
    upd(rz, vb + 2, br, ir); upd(rw, vb + 3, br, ir);
  };

  // Prologue: fill DEPTH-1 pipeline stages; prefetch q for tile 0.
  issue(0); issue(1); issue(2);
  float4 qv_cur = q4[w * CPW + ln];

  for (int i = 0; i < ITERS; ++i) {
    if (i + DEPTH - 1 < ITERS) issue(i + DEPTH - 1);
    // Prefetch q for the NEXT tile before any waits (LOADcnt != ASYNCcnt, so it
    // streams from L2 in the background while we wait on the async tile).
    float4 qv_next = qv_cur;
    if (i + 1 < ITERS) qv_next = q4[w * CPW + (i + 1) * 32 + ln];
    wait_async_tiles(ITERS - 1 - i);
    const int c = w * CPW + i * 32 + ln;
    const int buf = i & (DEPTH - 1);
    float4 tv = smem[w][buf][0][ln];
    float4 dv = smem[w][buf][1][ln];
    proc(tv, dv, qv_cur, c * 4);
    qv_cur = qv_next;
  }
  // tail: 8 chunks per wave via direct global b128 loads
  if (ln < TAIL) {
    const int c = w * CPW + ITERS * 32 + ln;
    proc(t4[c], d4[c], q4[c], c * 4);
  }

  // wave32 butterfly reduction with index tie-break
  for (int m = 16; m >= 1; m >>= 1) {
    float ov = __shfl_xor(bt, m, 32); int oi = __shfl_xor(it, m, 32);
    upd(ov, oi, bt, it);
    ov = __shfl_xor(br, m, 32); oi = __shfl_xor(ir, m, 32);
    upd(ov, oi, br, ir);
  }
  if (ln == 0) { s_val[0][w] = bt; s_idx[0][w] = it; s_val[1][w] = br; s_idx[1][w] = ir; }
  __syncthreads();
  if (tid == 0) {
    float v0 = s_val[0][0]; int i0 = s_idx[0][0];
    float v1 = s_val[1][0]; int i1 = s_idx[1][0];
    for (int k = 1; k < WAVES; ++k) {
      upd(s_val[0][k], s_idx[0][k], v0, i0);
      upd(s_val[1][k], s_idx[1][k], v1, i1);
    }
    targ_arg[row] = i0;
    recov[row]    = i1;
    const int dt  = dids[row];
    pd[row] = drow[dt];
    pt[row] = trow[dt];
  }
}

// ---------------- tiny per-batch sequential pass ----------------
__global__ __launch_bounds__(128) void finalize_kernel(
    const float* __restrict__ u, const float* __restrict__ temp,
    const int* __restrict__ dids, const int* __restrict__ bonus,
    const int* __restrict__ ndraft,
    const int* __restrict__ targ_arg, const int* __restrict__ recov,
    const float* __restrict__ pd, const float* __restrict__ pt,
    long long* __restrict__ out)
{
  const int b = blockIdx.x * blockDim.x + threadIdx.x;
  if (b >= B_) return;
  const bool greedy = (temp[b] == -1.0f);
  const int  n      = ndraft[b];

  long long o[L_ + 1];
  bool prefix = true;  // exclusive cumulative "all" of match/accept
  for (int l = 0; l < L_; ++l) {
    const int  idx   = b * L_ + l;
    const bool valid = l < n;
    const int  dt    = dids[idx];
    bool cond; int tok;
    if (greedy) {
      const int ta = targ_arg[idx];
      cond = (dt == ta) || !valid;
      tok  = ta;
    } else {
      const float pdv = pd[idx];
      const bool acc  = (pdv == 0.0f) || (pt[idx] >= u[idx] * pdv);
      cond = acc || !valid;
      tok  = acc ? dt : recov[idx];
    }
    const bool wr = valid && prefix;
    o[l] = wr ? (long long)tok : -1LL;
    prefix = prefix && cond;
  }
  o[L_] = -1LL;
  if (prefix) o[n] = (long long)bonus[b];  // full accept -> bonus at position n (1..8)

  for (int l = 0; l <= L_; ++l) out[b * (L_ + 1) + l] = o[l];
}

extern "C" void kernel_launch(void* const* d_in, const int* in_sizes, int n_in,
                              void* d_out, int out_size, void* d_ws, size_t ws_size,
                              hipStream_t stream) {
  (void)in_sizes; (void)n_in; (void)out_size; (void)ws_size;
  const float* draft  = (const float*)d_in[0];   // (B,L,V)
  const float* target = (const float*)d_in[1];   // (B,L,V)
  const float* u      = (const float*)d_in[2];   // (B,L)
  const float* q      = (const float*)d_in[3];   // (B,V)
  const float* temp   = (const float*)d_in[4];   // (B,)
  const int*   dids   = (const int*)d_in[5];     // (B,L)
  const int*   bonus  = (const int*)d_in[6];     // (B,)
  const int*   nd     = (const int*)d_in[7];     // (B,)

  int*   targ_arg = (int*)d_ws;                  // B*L ints
  int*   recov    = targ_arg + B_ * L_;          // B*L ints
  float* pd       = (float*)(recov + B_ * L_);   // B*L floats
  float* pt       = pd + B_ * L_;                // B*L floats
  long long* out  = (long long*)d_out;           // (B, L+1) int64

  row_reduce_kernel<<<B_ * L_, 256, 0, stream>>>(draft, target, q, dids,
                                                 targ_arg, recov, pd, pt);
  finalize_kernel<<<(B_ + 127) / 128, 128, 0, stream>>>(u, temp, dids, bonus, nd,
                                                        targ_arg, recov, pd, pt, out);
}